// FFTModule_69861938037263
// MI455X (gfx1250) — compile-verified
//
#include <hip/hip_runtime.h>
#include <math.h>

// ---------------------------------------------------------------------------
// FFTModule on MI455X (gfx1250): DFT-as-GEMM on f32 WMMA (16x16x4), with the
// B-operand panels staged into LDS by the Tensor Data Mover (TDM).
//   For each of 192 images (512x512):
//     Y = X * W          (row DFT;  W[k][n] = exp(-2*pi*i*k*n/512), symmetric)
//     Z = W * Y          (col DFT;  complex GEMM, 4 real chains)
//   mag = log(|Z|+1e-10) min-max normalized per image; phase -> [0,1].
// Workspace layout (floats): W_re | W_im | Y_re | Y_im | minmax(uint enc)
// Block = 256 threads = 8 waves; the 8 tiles share i0 and span 128 columns,
// so one TDM 2-D tile (128 wide x KCHUNK rows) serves the whole block.
// ---------------------------------------------------------------------------

typedef __attribute__((ext_vector_type(2))) float v2f;
typedef __attribute__((ext_vector_type(8))) float v8f;
typedef __attribute__((ext_vector_type(4))) unsigned int u32x4;
typedef __attribute__((ext_vector_type(8))) int i32x8;
typedef __attribute__((ext_vector_type(4))) int i32x4;

#define NN        512
#define NIMG      192                 // 64 batches * 3 channels
#define IMG_ELEMS (512 * 512)         // 262144
#define KCHUNK    64                  // K rows staged in LDS per TDM load
#define PANELW    128                 // columns per block panel
#define PI_F      3.14159265358979323846f

// Order-preserving float -> uint encoding (exact atomic min/max on floats).
__device__ __forceinline__ unsigned fenc(float f) {
    unsigned b = __float_as_uint(f);
    return (b & 0x80000000u) ? ~b : (b | 0x80000000u);
}
__device__ __forceinline__ float fdec(unsigned e) {
    unsigned b = (e & 0x80000000u) ? (e ^ 0x80000000u) : ~e;
    return __uint_as_float(b);
}

// ---------------------------------------------------------------------------
// TDM: DMA a 2-D tile (tile_w x tile_h f32, row stride = stride elems) from
// global memory into LDS at lds_byte_addr. D# layout per CDNA5 ISA 8.3-8.5.
// amdgpu-toolchain (clang-23) 6-arg builtin form:
//   (u32x4 g0, i32x8 g1, i32x4 g2, i32x4 g3, i32x8 extra, i32 cpol)
// ---------------------------------------------------------------------------
__device__ __forceinline__ void tdm_load_2d(unsigned int lds_byte_addr,
                                            const float* gptr,
                                            unsigned int tile_w,
                                            unsigned int tile_h,
                                            unsigned int stride) {
    const unsigned long long ga = (unsigned long long)(uintptr_t)gptr;
    u32x4 g0;
    g0[0] = 1u;                                                // count=1, user D#
    g0[1] = lds_byte_addr;                                     // lds_addr[31:0]
    g0[2] = (unsigned int)ga;                                  // global_addr[31:0]
    g0[3] = ((unsigned int)(ga >> 32) & 0x01FFFFFFu)           // global_addr[56:32]
            | 0x80000000u;                                     // type = 2 ("image")
    i32x8 g1;
    g1[0] = (int)0x00020000u;                  // wg_mask=0, data_size=2 (4 bytes)
    g1[1] = (int)(512u << 16);                 // tensor_dim0 = 512 (low 16 bits)
    g1[2] = (int)(512u << 16);                 // td0 hi = 0 | tensor_dim1 = 512 (lo)
    g1[3] = (int)(tile_w << 16);               // td1 hi = 0 | tile_dim0
    g1[4] = (int)tile_h;                       // tile_dim1 | tile_dim2 = 0
    g1[5] = (int)stride;                       // tensor_dim0_stride[31:0]
    g1[6] = 0;                                 // stride hi | tensor_dim1_stride lo
    g1[7] = 0;
    const i32x4 z4 = {0, 0, 0, 0};             // groups 2/3: 2-D tile, unused
    const i32x8 z8 = {0, 0, 0, 0, 0, 0, 0, 0};
    __builtin_amdgcn_tensor_load_to_lds(g0, g1, z4, z4, z8, 0);
}

// ---------------------------------------------------------------------------
// K0: twiddle matrix. theta folded mod 512 exactly in integer for precision.
// ---------------------------------------------------------------------------
__global__ void twiddle_init(float* __restrict__ wre, float* __restrict__ wim) {
    int idx = blockIdx.x * blockDim.x + threadIdx.x;   // 0 .. 262143
    int k = idx >> 9;
    int n = idx & 511;
    int t = (k * n) & 511;                             // exact (k*n) mod 512
    float theta = -2.0f * PI_F * (float)t * (1.0f / 512.0f);
    float s, c;
    __sincosf(theta, &s, &c);
    wre[idx] = c;
    wim[idx] = s;
}

// K0b: per-image min/max accumulators (encoded): min -> UINT_MAX, max -> 0.
__global__ void minmax_init(unsigned* __restrict__ mm) {
    for (int i = threadIdx.x; i < 2 * NIMG; i += blockDim.x)
        mm[i] = (i & 1) ? 0u : 0xFFFFFFFFu;            // even = min, odd = max
}

// ---------------------------------------------------------------------------
// Stage A: Y = X * W. One 16x16 tile per wave; block-shared B panels in LDS.
// Fragment layout per CDNA5 ISA 7.12.2:
//   A 16x4:  lane = hi*16+m holds A[i0+m][k+2*hi], A[i0+m][k+2*hi+1]
//   B 4x16:  lane = hi*16+m holds B[k+2*hi][j0+m], B[k+2*hi+1][j0+m]
//   C 16x16: vgpr r, lane hi*16+m -> C[i0 + r + 8*hi][j0 + m]
// ---------------------------------------------------------------------------
__global__ void __launch_bounds__(256) row_dft(
        const float* __restrict__ x,
        const float* __restrict__ wre, const float* __restrict__ wim,
        float* __restrict__ yre, float* __restrict__ yim) {
    __shared__ float ldsRe[KCHUNK * PANELW];           // 32 KB
    __shared__ float ldsIm[KCHUNK * PANELW];           // 32 KB

    const int wave = threadIdx.x >> 5;
    const int lane = threadIdx.x & 31;
    const int hi   = lane >> 4;
    const int m    = lane & 15;
    const int k2   = 2 * hi;
    const int img  = blockIdx.y;
    const int t    = blockIdx.x * 8 + wave;            // 0..1023 tiles/image
    const int i0   = (t >> 5) << 4;                    // shared by all 8 waves
    const int jblk = (blockIdx.x & 3) * PANELW;        // block's column panel
    const int jl   = wave * 16;                        // wave's cols in panel
    const int j0   = jblk + jl;

    const unsigned int ldsReA = (unsigned int)(uintptr_t)(void*)ldsRe;
    const unsigned int ldsImA = (unsigned int)(uintptr_t)(void*)ldsIm;

    const size_t ioff = (size_t)img * IMG_ELEMS;
    const float* X = x + ioff;

    v8f cre = {};
    v8f cim = {};

    const int arow = (i0 + m) * NN;
    for (int kc = 0; kc < NN; kc += KCHUNK) {
        __syncthreads();                               // panel reuse fence
        if (wave == 0) {                               // TDM is wave-level
            tdm_load_2d(ldsReA, wre + (size_t)kc * NN + jblk, PANELW, KCHUNK, NN);
            tdm_load_2d(ldsImA, wim + (size_t)kc * NN + jblk, PANELW, KCHUNK, NN);
            __builtin_amdgcn_s_wait_tensorcnt(0);
        }
        __syncthreads();

        for (int k = 0; k < KCHUNK; k += 4) {
            __builtin_prefetch(&X[arow + kc + k + 16], 0, 1);
            v2f a, bre, bim;
            a.x   = X[arow + kc + k + k2];
            a.y   = X[arow + kc + k + k2 + 1];
            bre.x = ldsRe[(k + k2) * PANELW + jl + m];
            bre.y = ldsRe[(k + k2 + 1) * PANELW + jl + m];
            bim.x = ldsIm[(k + k2) * PANELW + jl + m];
            bim.y = ldsIm[(k + k2 + 1) * PANELW + jl + m];
            cre = __builtin_amdgcn_wmma_f32_16x16x4_f32(false, a, false, bre,
                                                        (short)0, cre, false, false);
            cim = __builtin_amdgcn_wmma_f32_16x16x4_f32(false, a, false, bim,
                                                        (short)0, cim, false, false);
        }
    }

#pragma unroll
    for (int r = 0; r < 8; ++r) {
        const int row = i0 + r + 8 * hi;
        const int col = j0 + m;
        yre[ioff + row * NN + col] = cre[r];
        yim[ioff + row * NN + col] = cim[r];
    }
}

// ---------------------------------------------------------------------------
// Stage B: Z = W * Y (complex GEMM, 4 WMMA chains; Y panels TDM-staged in
// LDS) fused with the pointwise epilogue: log-magnitude, phase remap, output
// stores, per-image min/max. f32 WMMA has no A-neg modifier (ISA 7.12: NEG =
// C-neg only) so -W_im is negated in VALU registers for the second Zre chain.
// ---------------------------------------------------------------------------
__global__ void __launch_bounds__(256) col_dft_pointwise(
        const float* __restrict__ wre, const float* __restrict__ wim,
        const float* __restrict__ yre, const float* __restrict__ yim,
        float* __restrict__ out, unsigned* __restrict__ mm) {
    __shared__ float ldsRe[KCHUNK * PANELW];           // 32 KB
    __shared__ float ldsIm[KCHUNK * PANELW];           // 32 KB
    __shared__ float smin[256];
    __shared__ float smax[256];

    const int wave = threadIdx.x >> 5;
    const int lane = threadIdx.x & 31;
    const int hi   = lane >> 4;
    const int m    = lane & 15;
    const int k2   = 2 * hi;
    const int img  = blockIdx.y;
    const int t    = blockIdx.x * 8 + wave;
    const int i0   = (t >> 5) << 4;
    const int jblk = (blockIdx.x & 3) * PANELW;
    const int jl   = wave * 16;
    const int j0   = jblk + jl;

    const unsigned int ldsReA = (unsigned int)(uintptr_t)(void*)ldsRe;
    const unsigned int ldsImA = (unsigned int)(uintptr_t)(void*)ldsIm;

    const size_t ioff = (size_t)img * IMG_ELEMS;

    v8f zre = {};
    v8f zim = {};

    const int arow = (i0 + m) * NN;
    for (int kc = 0; kc < NN; kc += KCHUNK) {
        __syncthreads();
        if (wave == 0) {
            tdm_load_2d(ldsReA, yre + ioff + (size_t)kc * NN + jblk, PANELW, KCHUNK, NN);
            tdm_load_2d(ldsImA, yim + ioff + (size_t)kc * NN + jblk, PANELW, KCHUNK, NN);
            __builtin_amdgcn_s_wait_tensorcnt(0);
        }
        __syncthreads();

        for (int k = 0; k < KCHUNK; k += 4) {
            __builtin_prefetch(&wre[arow + kc + k + 16], 0, 1);
            v2f awre, awim, byre, byim;
            awre.x = wre[arow + kc + k + k2];
            awre.y = wre[arow + kc + k + k2 + 1];
            awim.x = wim[arow + kc + k + k2];
            awim.y = wim[arow + kc + k + k2 + 1];
            byre.x = ldsRe[(k + k2) * PANELW + jl + m];
            byre.y = ldsRe[(k + k2 + 1) * PANELW + jl + m];
            byim.x = ldsIm[(k + k2) * PANELW + jl + m];
            byim.y = ldsIm[(k + k2 + 1) * PANELW + jl + m];

            zre = __builtin_amdgcn_wmma_f32_16x16x4_f32(false, awre, false, byre,
                                                        (short)0, zre, false, false);
            v2f awim_n;
            awim_n.x = -awim.x;
            awim_n.y = -awim.y;
            zre = __builtin_amdgcn_wmma_f32_16x16x4_f32(false, awim_n, false, byim,
                                                        (short)0, zre, false, false);
            zim = __builtin_amdgcn_wmma_f32_16x16x4_f32(false, awre, false, byim,
                                                        (short)0, zim, false, false);
            zim = __builtin_amdgcn_wmma_f32_16x16x4_f32(false, awim, false, byre,
                                                        (short)0, zim, false, false);
        }
    }

    // Epilogue: mag/phase + local min/max over this wave's 8 elements.
    const int b  = img / 3;
    const int ch = img % 3;
    const size_t magbase = (size_t)(b * 6 + ch) * IMG_ELEMS;
    const size_t phbase  = (size_t)(b * 6 + 3 + ch) * IMG_ELEMS;

    float lmin = 3.402823466e38f;
    float lmax = -3.402823466e38f;
#pragma unroll
    for (int r = 0; r < 8; ++r) {
        const float re = zre[r];
        const float im = zim[r];
        const float mag = __logf(sqrtf(re * re + im * im) + 1e-10f);
        const float ph  = (atan2f(im, re) + PI_F) * (0.5f / PI_F);
        const int row = i0 + r + 8 * hi;
        const int col = j0 + m;
        out[magbase + row * NN + col] = mag;
        out[phbase + row * NN + col]  = ph;
        lmin = fminf(lmin, mag);
        lmax = fmaxf(lmax, mag);
    }

    // Block reduction (whole block belongs to one image), then one atomic pair.
    smin[threadIdx.x] = lmin;
    smax[threadIdx.x] = lmax;
    __syncthreads();
    for (int s = 128; s > 0; s >>= 1) {
        if (threadIdx.x < s) {
            smin[threadIdx.x] = fminf(smin[threadIdx.x], smin[threadIdx.x + s]);
            smax[threadIdx.x] = fmaxf(smax[threadIdx.x], smax[threadIdx.x + s]);
        }
        __syncthreads();
    }
    if (threadIdx.x == 0) {
        atomicMin(&mm[img * 2 + 0], fenc(smin[0]));
        atomicMax(&mm[img * 2 + 1], fenc(smax[0]));
    }
}

// ---------------------------------------------------------------------------
// K4: per-image min-max normalization of the magnitude planes (in d_out).
// ---------------------------------------------------------------------------
__global__ void normalize_mag(float* __restrict__ out,
                              const unsigned* __restrict__ mm) {
    const int idx = blockIdx.x * blockDim.x + threadIdx.x;
    if (idx >= NIMG * IMG_ELEMS) return;
    const int img = idx >> 18;                 // / IMG_ELEMS
    const int within = idx & (IMG_ELEMS - 1);
    const float mn = fdec(mm[img * 2 + 0]);
    const float mx = fdec(mm[img * 2 + 1]);
    const float denom = mx - mn;
    const int b  = img / 3;
    const int ch = img % 3;
    const size_t o = (size_t)(b * 6 + ch) * IMG_ELEMS + within;
    const float v = out[o];
    out[o] = (denom > 0.0f) ? (v - mn) / denom : v;
}

// ---------------------------------------------------------------------------
extern "C" void kernel_launch(void* const* d_in, const int* in_sizes, int n_in,
                              void* d_out, int out_size, void* d_ws, size_t ws_size,
                              hipStream_t stream) {
    (void)in_sizes; (void)n_in; (void)out_size; (void)ws_size;

    const float* x = (const float*)d_in[0];
    float* out = (float*)d_out;

    float* ws  = (float*)d_ws;
    float* wre = ws;
    float* wim = ws + IMG_ELEMS;
    float* yre = ws + 2 * (size_t)IMG_ELEMS;
    float* yim = yre + (size_t)NIMG * IMG_ELEMS;
    unsigned* mm = (unsigned*)(yim + (size_t)NIMG * IMG_ELEMS);

    twiddle_init<<<IMG_ELEMS / 256, 256, 0, stream>>>(wre, wim);
    minmax_init<<<1, 256, 0, stream>>>(mm);

    dim3 grid(128, NIMG);                        // 8 tiles/block * 128 = 1024 tiles/image
    row_dft<<<grid, 256, 0, stream>>>(x, wre, wim, yre, yim);
    col_dft_pointwise<<<grid, 256, 0, stream>>>(wre, wim, yre, yim, out, mm);

    const int total = NIMG * IMG_ELEMS;
    normalize_mag<<<(total + 255) / 256, 256, 0, stream>>>(out, mm);
}